// D2D_69363721830777
// MI455X (gfx1250) — compile-verified
//
#include <hip/hip_runtime.h>
#include <cstdint>

// Problem constants (reference: des (B=2, D=64, H=384, W=384) fp32, K=5, dil=4)
#define DCH    64
#define HH     384
#define WW     384
#define TILE_W 64          // center pixels per workgroup (4 waves x 16)
#define NCOLS  80          // neighbor strip: TILE_W + 2*8 (dx in [-8,8])
#define NWAVES 4

typedef float v2f __attribute__((ext_vector_type(2)));
typedef float v8f __attribute__((ext_vector_type(8)));
typedef unsigned int v4u __attribute__((ext_vector_type(4)));
typedef int  v8i __attribute__((ext_vector_type(8)));
typedef int  v4i __attribute__((ext_vector_type(4)));

#define HAS_TDM __has_builtin(__builtin_amdgcn_tensor_load_to_lds)

__device__ __forceinline__ uint32_t lds_off(const void* p) {
  // generic pointer to LDS: low 32 bits are the LDS byte offset (aperture in high bits)
  return (uint32_t)(uintptr_t)p;
}

// Per-element async global->LDS copy (clamped edge path). ASYNCcnt-tracked.
__device__ __forceinline__ void async_b32(const float* g, float* l) {
  asm volatile("global_load_async_to_lds_b32 %0, %1, off"
               :: "v"(lds_off(l)), "v"(g) : "memory");
}

__device__ __forceinline__ void wait_async0() {
#if __has_builtin(__builtin_amdgcn_s_wait_asynccnt)
  __builtin_amdgcn_s_wait_asynccnt(0);
#else
  asm volatile("s_wait_asynccnt 0x0" ::: "memory");
#endif
}

#if HAS_TDM
// One TDM descriptor: 2D tile 80 cols (contiguous x) x 64 channels (stride H*W),
// fp32 elements, into LDS at `ldst`.
__device__ __forceinline__ void tdm_row_load(const float* g, float* ldst) {
  uint64_t ga = (uint64_t)(uintptr_t)g;
  v4u g0;
  g0[0] = 1u;                                          // count=1 (valid), no gather
  g0[1] = lds_off(ldst);                               // lds_addr
  g0[2] = (uint32_t)(ga & 0xffffffffu);                // global_addr[31:0]
  g0[3] = (uint32_t)((ga >> 32) & 0x1ffffffu)          // global_addr[56:32]
        | (2u << 30);                                  // type=2 ("image")
  const uint32_t d0s = (uint32_t)WW;                   // tensor_dim0_stride
  const uint32_t d1s = (uint32_t)(HH * WW);            // tensor_dim1_stride
  v8i g1;
  g1[0] = (int)(2u << 16);                             // data_size=2 (4 bytes)
  g1[1] = (int)(((uint32_t)NCOLS & 0xffffu) << 16);    // tensor_dim0 lo16
  g1[2] = (int)((((uint32_t)NCOLS >> 16) & 0xffffu)    // tensor_dim0 hi16
        | (((uint32_t)DCH & 0xffffu) << 16));          // tensor_dim1 lo16
  g1[3] = (int)((((uint32_t)DCH >> 16) & 0xffffu)      // tensor_dim1 hi16
        | (((uint32_t)NCOLS & 0xffffu) << 16));        // tile_dim0 = 80
  g1[4] = (int)((uint32_t)DCH & 0xffffu);              // tile_dim1 = 64, tile_dim2 = 0
  g1[5] = (int)d0s;                                    // dim0 stride lo32
  g1[6] = (int)((d1s & 0xffffu) << 16);                // dim0 stride hi16 (=0) | dim1 stride lo16
  g1[7] = (int)(d1s >> 16);                            // dim1 stride [47:16]
  v4i g2 = {0, 0, 0, 0};
  v4i g3 = {0, 0, 0, 0};
  v8i g4 = {0, 0, 0, 0, 0, 0, 0, 0};                   // 6-arg toolchain: extra group
  __builtin_amdgcn_tensor_load_to_lds(g0, g1, g2, g3, g4, 0);
}
#endif

__global__ __launch_bounds__(128) void nss_kernel(const float* __restrict__ des,
                                                  float* __restrict__ out) {
  __shared__ float buf[2][DCH * NCOLS];     // double-buffered [ch][col] strips
  __shared__ float nrm[NCOLS];              // per-row neighbor squared norms
  __shared__ float cnrm[TILE_W];            // center squared norms
  __shared__ float csum[TILE_W];            // center channel sums (for s_abs)
  __shared__ float scratch[NWAVES][16 * 32];// per-wave WMMA C extraction

  const int tid  = threadIdx.x;
  const int wave = tid >> 5;
  const int lane = tid & 31;
  const int m    = lane & 15;
  const int hi   = lane >> 4;

  const int x0 = blockIdx.x * TILE_W;
  const int y  = blockIdx.y;
  const int b  = blockIdx.z;

  const bool interior = (x0 >= 8) && (x0 + NCOLS - 8 <= WW);
  const size_t plane = (size_t)HH * WW;
  const float* base = des + (size_t)b * DCH * plane;

  const int ord[5] = {0, -8, -4, 4, 8};     // dy order: center row first

  auto fill_row = [&](int i, int slot) {
    int yy = y + ord[i];
    yy = yy < 0 ? 0 : (yy >= HH ? HH - 1 : yy);
    const float* rowp = base + (size_t)yy * WW;
#if HAS_TDM
    if (interior) {
      if (wave == 0) tdm_row_load(rowp + (x0 - 8), &buf[slot][0]);
      return;
    }
#endif
    for (int it = 0; it < (DCH * NCOLS) / 128; ++it) {
      int idx = it * 128 + tid;
      int ch = idx / NCOLS;
      int c  = idx - ch * NCOLS;
      int xx = x0 - 8 + c;
      xx = xx < 0 ? 0 : (xx >= WW ? WW - 1 : xx);
      async_b32(rowp + (size_t)ch * plane + xx, &buf[slot][ch * NCOLS + c]);
    }
  };

  auto wait_fill = [&]() {
    wait_async0();
#if HAS_TDM
    if (wave == 0) __builtin_amdgcn_s_wait_tensorcnt(0);
#endif
    __syncthreads();
  };

  v2f areg[16];                 // A-matrix: 16 center pixels x 64 ch, resident in VGPRs
  float acc = 0.0f;             // per-pixel L2 accumulation (lanes 0..15)

  fill_row(0, 0);
  wait_fill();

  for (int i = 0; i < 5; ++i) {
    const int cur = i & 1;
    const int nxt = cur ^ 1;
    float* tb = &buf[cur][0];

    // ---- stage 1: per-strip squared norms (+ center stats, A regs on first row)
    for (int c = tid; c < NCOLS; c += 128) {
      float s2 = 0.0f;
      for (int ch = 0; ch < DCH; ++ch) { float v = tb[ch * NCOLS + c]; s2 += v * v; }
      nrm[c] = s2;
    }
    if (i == 0) {
      for (int c = tid; c < TILE_W; c += 128) {
        float s1 = 0.0f, s2 = 0.0f;
        for (int ch = 0; ch < DCH; ++ch) {
          float v = tb[ch * NCOLS + 8 + c];
          s1 += v; s2 += v * v;
        }
        csum[c] = s1; cnrm[c] = s2;
      }
#pragma unroll
      for (int kq = 0; kq < 16; ++kq) {
        const int ch = kq * 4 + 2 * hi;
        const int cc = 8 + wave * 16 + m;
        v2f a;
        a.x = tb[ch * NCOLS + cc];
        a.y = tb[(ch + 1) * NCOLS + cc];
        areg[kq] = a;
      }
    }

    // ---- stage 2: kick off DMA of next dy strip into the other buffer
    if (i + 1 < 5) fill_row(i + 1, nxt);

    __syncthreads();  // norms/center stats visible

    // ---- stage 3: 16x32 dot block = two independent chains of
    //      V_WMMA_F32_16X16X4_F32, B operands prefetched to registers
    float* sc = &scratch[wave][0];
    {
      v2f b0[16], b1[16];
      const int cb0 = wave * 16;        // half 0: cols [w*16, w*16+16)
      const int cb1 = wave * 16 + 16;   // half 1: cols [w*16+16, w*16+32)
#pragma unroll
      for (int kq = 0; kq < 16; ++kq) {
        const int ch = kq * 4 + 2 * hi;
        b0[kq].x = tb[ch * NCOLS + cb0 + m];
        b0[kq].y = tb[(ch + 1) * NCOLS + cb0 + m];
        b1[kq].x = tb[ch * NCOLS + cb1 + m];
        b1[kq].y = tb[(ch + 1) * NCOLS + cb1 + m];
      }
      v8f c0 = {0.f, 0.f, 0.f, 0.f, 0.f, 0.f, 0.f, 0.f};
      v8f c1 = {0.f, 0.f, 0.f, 0.f, 0.f, 0.f, 0.f, 0.f};
#pragma unroll
      for (int kq = 0; kq < 16; ++kq) {
        c0 = __builtin_amdgcn_wmma_f32_16x16x4_f32(
            false, areg[kq], false, b0[kq], (short)0, c0, false, false);
        c1 = __builtin_amdgcn_wmma_f32_16x16x4_f32(
            false, areg[kq], false, b1[kq], (short)0, c1, false, false);
      }
#pragma unroll
      for (int rr = 0; rr < 8; ++rr) {
        sc[(rr + 8 * hi) * 32 + m]      = c0[rr];
        sc[(rr + 8 * hi) * 32 + 16 + m] = c1[rr];
      }
    }

    // ---- stage 4: pull the 5 dx-diagonals, combine with norms, accumulate sqrt
    if (lane < 16) {
      const int ci = wave * 16 + lane;
      const float nc = cnrm[ci];
#pragma unroll
      for (int o = 0; o < 5; ++o) {
        const int j = lane + 4 * o;                 // cols i, i+4, ..., i+16
        const float d  = sc[lane * 32 + j];
        const float nq = nrm[wave * 16 + j];
        float d2 = nc + nq - 2.0f * d;
        acc += sqrtf(d2 > 0.0f ? d2 : 0.0f);
      }
    }

    // ---- stage 5: fence DMA + release buffers for reuse
    if (i + 1 < 5) wait_fill();
  }

  if (lane < 16) {
    const int ci = wave * 16 + lane;
    const float mu  = csum[ci] * (1.0f / DCH);
    float var = cnrm[ci] * (1.0f / DCH) - mu * mu;
    var = var > 0.0f ? var : 0.0f;
    const float s_abs = sqrtf(var);
    const float s_rel = acc * (1.0f / 25.0f);
    const float s_raw = s_abs * s_rel;
    out[((size_t)b * HH + y) * WW + (x0 + ci)] = s_raw / (s_raw + 1.0f);
  }
}

extern "C" void kernel_launch(void* const* d_in, const int* in_sizes, int n_in,
                              void* d_out, int out_size, void* d_ws, size_t ws_size,
                              hipStream_t stream) {
  const float* des = (const float*)d_in[0];
  float* out = (float*)d_out;
  const int Bn = in_sizes[0] / (DCH * HH * WW);
  dim3 grid(WW / TILE_W, HH, Bn);
  nss_kernel<<<grid, 128, 0, stream>>>(des, out);
}